// Attention_13907104104901
// MI455X (gfx1250) — compile-verified
//
#include <hip/hip_runtime.h>
#include <hip/hip_bf16.h>

#define Bdim 8
#define Sdim 512
#define Hdim 768
#define JT 4   // j-rows per score block

typedef __attribute__((ext_vector_type(2))) float v2f;
typedef __attribute__((ext_vector_type(8))) float v8f;

// gfx1250 hardware v_tanh_f32 (TRANS unit) when available; otherwise exp2/rcp.
__device__ __forceinline__ float fast_tanh(float x) {
#if __has_builtin(__builtin_amdgcn_tanhf)
  return __builtin_amdgcn_tanhf(x);
#else
  x = fminf(fmaxf(x, -10.0f), 10.0f);
  float e = __builtin_amdgcn_exp2f(x * -2.8853900817779268f);   // e^{-2x}
  return (1.0f - e) * __builtin_amdgcn_rcpf(1.0f + e);
#endif
}

// ---------------------------------------------------------------------------
// Kernel 1: P[sel][r,o] = sum_h A[r,h]*W[sel][o,h] + bias[sel][o], sel={Wa,Ua}
// A = outputs as [B*S, H].  One wave per 32x32 output tile (2x2 WMMA register
// blocking): per K-step of 4 -> 4 v2f loads feed 4 V_WMMA_F32_16X16X4_F32.
//
// Fragment layout (ISA 7.12.2, 32-bit 16x4 A):
//   lane L<16 : a = {A[M=L][kk+0], A[M=L][kk+1]}
//   lane L>=16: a = {A[M=L-16][kk+2], A[M=L-16][kk+3]}
// B mirrored; C/D: VGPR r -> M=r (lanes 0-15), M=r+8 (lanes 16-31).
// ---------------------------------------------------------------------------
__global__ __launch_bounds__(256) void proj_gemm(
    const float* __restrict__ A,
    const float* __restrict__ W0, const float* __restrict__ b0,
    const float* __restrict__ W1, const float* __restrict__ b1,
    float* __restrict__ P0, float* __restrict__ P1) {
  const int wave = blockIdx.x * 8 + (threadIdx.x >> 5);
  const int lane = threadIdx.x & 31;
  const int lr = lane & 15;
  const int hi = lane >> 4;

  const int nTiles = Hdim / 32;                         // 24
  const int tilesPerMat = (Bdim * Sdim / 32) * nTiles;  // 128*24
  const int sel = wave / tilesPerMat;
  int rem = wave - sel * tilesPerMat;
  const int mT = rem / nTiles;
  const int nT = rem - mT * nTiles;
  const int row0 = mT * 32;
  const int col0 = nT * 32;

  const float* W = sel ? W1 : W0;
  const float* bias = sel ? b1 : b0;
  float* P = sel ? P1 : P0;

  const float* aP0 = A + (size_t)(row0 + lr) * Hdim + 2 * hi;
  const float* aP1 = aP0 + (size_t)16 * Hdim;
  const float* bP0 = W + (size_t)(col0 + lr) * Hdim + 2 * hi;
  const float* bP1 = bP0 + (size_t)16 * Hdim;

  v8f c00 = {}, c01 = {}, c10 = {}, c11 = {};
#pragma unroll 2
  for (int kk = 0; kk < Hdim; kk += 4) {
    v2f a0 = *(const v2f*)(aP0 + kk);
    v2f a1 = *(const v2f*)(aP1 + kk);
    v2f bb0 = *(const v2f*)(bP0 + kk);
    v2f bb1 = *(const v2f*)(bP1 + kk);
    c00 = __builtin_amdgcn_wmma_f32_16x16x4_f32(false, a0, false, bb0, (short)0, c00, false, false);
    c01 = __builtin_amdgcn_wmma_f32_16x16x4_f32(false, a0, false, bb1, (short)0, c01, false, false);
    c10 = __builtin_amdgcn_wmma_f32_16x16x4_f32(false, a1, false, bb0, (short)0, c10, false, false);
    c11 = __builtin_amdgcn_wmma_f32_16x16x4_f32(false, a1, false, bb1, (short)0, c11, false, false);
  }

  const float bv0 = bias[col0 + lr];
  const float bv1 = bias[col0 + 16 + lr];
#pragma unroll
  for (int r = 0; r < 8; ++r) {
    const size_t rowA = (size_t)(row0 + r + 8 * hi) * Hdim;
    const size_t rowB = (size_t)(row0 + 16 + r + 8 * hi) * Hdim;
    P[rowA + col0 + lr]      = c00[r] + bv0;
    P[rowA + col0 + 16 + lr] = c01[r] + bv1;
    P[rowB + col0 + lr]      = c10[r] + bv0;
    P[rowB + col0 + 16 + lr] = c11[r] + bv1;
  }
}

// ---------------------------------------------------------------------------
// Kernel 2: JT=4 rows (b, j0..j0+3) per 256-thread block.
//   sc[j,k] = sum_h tanh(Wq[start+j,h] + Uk[k,h]) * va[h] + vb,  k < start
//   attn[b,j,k] = masked softmax; rows j >= S-start are all zero.
// Each global Uk float4 feeds 16 TRANS tanh + 16 VALU fma (4 j-rows x 4 comps)
// -> TRANS-bound instead of L2-bound.  j-tile validity is block-uniform.
// ---------------------------------------------------------------------------
__global__ __launch_bounds__(256) void score_softmax(
    const float* __restrict__ Wq, const float* __restrict__ Uk,
    const float* __restrict__ va_g, const float* __restrict__ vb_g,
    const int* __restrict__ mask, float* __restrict__ attn) {
  __shared__ __align__(16) float wq_s[JT][Hdim];
  __shared__ __align__(16) float va_s[Hdim];
  __shared__ float red[256];

  const int tid = threadIdx.x;
  const int jTiles = Sdim / JT;                // 128
  const int b = blockIdx.x / jTiles;
  const int j0 = (blockIdx.x - b * jTiles) * JT;
  const int start = mask[2 * b];
  const int jEnd = Sdim - start;               // rows j < jEnd are valid
  float* arow0 = attn + ((size_t)b * Sdim + j0) * Sdim;

  if (j0 >= jEnd) {                            // whole tile zero
#pragma unroll
    for (int jj = 0; jj < JT; ++jj)
      for (int k = tid; k < Sdim; k += 256) arow0[(size_t)jj * Sdim + k] = 0.0f;
    return;
  }

  for (int i = tid; i < Hdim; i += 256) va_s[i] = va_g[i];
#pragma unroll
  for (int jj = 0; jj < JT; ++jj) {
    if (j0 + jj < jEnd) {
      const float* wqrow = Wq + ((size_t)b * Sdim + start + j0 + jj) * Hdim;
      for (int i = tid; i < Hdim; i += 256) wq_s[jj][i] = wqrow[i];
    } else {
      for (int i = tid; i < Hdim; i += 256) wq_s[jj][i] = 0.0f;
    }
  }
  __syncthreads();

  const float vb = vb_g[0];
  float sc[2][JT];

#pragma unroll
  for (int s2 = 0; s2 < 2; ++s2) {
    const int k = tid + 256 * s2;
#pragma unroll
    for (int jj = 0; jj < JT; ++jj) sc[s2][jj] = -3.0e38f;
    if (k < start) {
      const float4* up = (const float4*)(Uk + ((size_t)b * Sdim + k) * Hdim);
      const float4* vp = (const float4*)va_s;
      float acc[JT] = {0.0f, 0.0f, 0.0f, 0.0f};
#pragma unroll 2
      for (int h = 0; h < Hdim / 4; ++h) {
        const float4 u = up[h];
        const float4 v = vp[h];
#pragma unroll
        for (int jj = 0; jj < JT; ++jj) {
          const float4 w = ((const float4*)wq_s[jj])[h];
          acc[jj] = fmaf(fast_tanh(w.x + u.x), v.x, acc[jj]);
          acc[jj] = fmaf(fast_tanh(w.y + u.y), v.y, acc[jj]);
          acc[jj] = fmaf(fast_tanh(w.z + u.z), v.z, acc[jj]);
          acc[jj] = fmaf(fast_tanh(w.w + u.w), v.w, acc[jj]);
        }
      }
#pragma unroll
      for (int jj = 0; jj < JT; ++jj) sc[s2][jj] = acc[jj] + vb;
    }
  }

  const int nj = (jEnd - j0 < JT) ? (jEnd - j0) : JT;   // block-uniform
  for (int jj = 0; jj < JT; ++jj) {
    float* arow = arow0 + (size_t)jj * Sdim;
    if (jj >= nj) {                                     // uniform branch
      arow[tid] = 0.0f;
      arow[tid + 256] = 0.0f;
      continue;
    }
    // block max over valid k (invalid slots hold -3e38)
    red[tid] = fmaxf(sc[0][jj], sc[1][jj]);
    __syncthreads();
    for (int off = 128; off > 0; off >>= 1) {
      if (tid < off) red[tid] = fmaxf(red[tid], red[tid + off]);
      __syncthreads();
    }
    const float gmax = red[0];
    __syncthreads();
    // block sum of exp
    float e0 = 0.0f, e1 = 0.0f, lsum = 0.0f;
    if (tid < start)       { e0 = __expf(sc[0][jj] - gmax); lsum += e0; }
    if (tid + 256 < start) { e1 = __expf(sc[1][jj] - gmax); lsum += e1; }
    red[tid] = lsum;
    __syncthreads();
    for (int off = 128; off > 0; off >>= 1) {
      if (tid < off) red[tid] += red[tid + off];
      __syncthreads();
    }
    const float inv = 1.0f / red[0];
    __syncthreads();
    arow[tid]       = (tid < start)       ? e0 * inv : 0.0f;
    arow[tid + 256] = (tid + 256 < start) ? e1 * inv : 0.0f;
  }
}

// ---------------------------------------------------------------------------
// Kernel 3: contexts[b] = attn[b] (SxS) @ outputs[b] (SxH), fp32 WMMA with
// 2x2 register blocking (32x32 per wave).  K truncated to start; zero-row
// tiles short-circuited with wave-uniform branches (EXEC all-ones for WMMA).
// ---------------------------------------------------------------------------
__global__ __launch_bounds__(256) void context_gemm(
    const float* __restrict__ attn, const float* __restrict__ outs,
    const int* __restrict__ mask, float* __restrict__ ctx) {
  const int wave = blockIdx.x * 8 + (threadIdx.x >> 5);
  const int lane = threadIdx.x & 31;
  const int lr = lane & 15;
  const int hi = lane >> 4;

  const int nTiles = Hdim / 32;                 // 24
  const int tilesPerB = (Sdim / 32) * nTiles;   // 16*24
  const int b = wave / tilesPerB;
  int rem = wave - b * tilesPerB;
  const int mT = rem / nTiles;
  const int nT = rem - mT * nTiles;
  const int row0 = mT * 32;
  const int col0 = nT * 32;
  const int start = mask[2 * b];

  float* cOut = ctx + (size_t)b * Sdim * Hdim;

  if (row0 >= Sdim - start) {                   // whole 32-row tile zero
#pragma unroll
    for (int r = 0; r < 8; ++r) {
      const size_t rowA = (size_t)(row0 + r + 8 * hi) * Hdim;
      const size_t rowB = (size_t)(row0 + 16 + r + 8 * hi) * Hdim;
      cOut[rowA + col0 + lr] = 0.0f;
      cOut[rowA + col0 + 16 + lr] = 0.0f;
      cOut[rowB + col0 + lr] = 0.0f;
      cOut[rowB + col0 + 16 + lr] = 0.0f;
    }
    return;
  }

  const float* Ab = attn + (size_t)b * Sdim * Sdim;
  const float* Bb = outs + (size_t)b * Sdim * Hdim;
  const int kmax = (start + 3) & ~3;            // attn cols >= start are zero

  const float* aP0 = Ab + (size_t)(row0 + lr) * Sdim + 2 * hi;
  const float* aP1 = aP0 + (size_t)16 * Sdim;
  const float* bC0 = Bb + col0 + lr;
  const float* bC1 = Bb + col0 + 16 + lr;

  v8f c00 = {}, c01 = {}, c10 = {}, c11 = {};
#pragma unroll 2
  for (int kk = 0; kk < kmax; kk += 4) {
    v2f a0 = *(const v2f*)(aP0 + kk);
    v2f a1 = *(const v2f*)(aP1 + kk);
    const size_t r0 = (size_t)(kk + 2 * hi) * Hdim;
    const size_t r1 = (size_t)(kk + 2 * hi + 1) * Hdim;
    v2f bb0, bb1;
    bb0.x = bC0[r0]; bb0.y = bC0[r1];
    bb1.x = bC1[r0]; bb1.y = bC1[r1];
    c00 = __builtin_amdgcn_wmma_f32_16x16x4_f32(false, a0, false, bb0, (short)0, c00, false, false);
    c01 = __builtin_amdgcn_wmma_f32_16x16x4_f32(false, a0, false, bb1, (short)0, c01, false, false);
    c10 = __builtin_amdgcn_wmma_f32_16x16x4_f32(false, a1, false, bb0, (short)0, c10, false, false);
    c11 = __builtin_amdgcn_wmma_f32_16x16x4_f32(false, a1, false, bb1, (short)0, c11, false, false);
  }
#pragma unroll
  for (int r = 0; r < 8; ++r) {
    const size_t rowA = (size_t)(row0 + r + 8 * hi) * Hdim;
    const size_t rowB = (size_t)(row0 + 16 + r + 8 * hi) * Hdim;
    cOut[rowA + col0 + lr]      = c00[r];
    cOut[rowA + col0 + 16 + lr] = c01[r];
    cOut[rowB + col0 + lr]      = c10[r];
    cOut[rowB + col0 + 16 + lr] = c11[r];
  }
}

// ---------------------------------------------------------------------------
extern "C" void kernel_launch(void* const* d_in, const int* in_sizes, int n_in,
                              void* d_out, int out_size, void* d_ws, size_t ws_size,
                              hipStream_t stream) {
  const float* outputs = (const float*)d_in[0];   // (B,S,H) fp32
  const int*   mask    = (const int*)d_in[1];     // (B,2) int32, [:,0] = start
  const float* Wa_w    = (const float*)d_in[2];   // (H,H)
  const float* Wa_b    = (const float*)d_in[3];   // (H,)
  const float* Ua_w    = (const float*)d_in[4];   // (H,H)
  const float* Ua_b    = (const float*)d_in[5];   // (H,)
  const float* Va_w    = (const float*)d_in[6];   // (1,H)
  const float* Va_b    = (const float*)d_in[7];   // (1,)

  float* attnOut = (float*)d_out;                       // B*S*S
  float* ctxOut  = attnOut + (size_t)Bdim * Sdim * Sdim;

  float* Wq = (float*)d_ws;                             // [B*S, H]
  float* Uk = Wq + (size_t)Bdim * Sdim * Hdim;          // [B*S, H]

  // Kernel 1: 2 * (4096/32)*(768/32) = 6144 waves -> 768 blocks of 8 waves
  proj_gemm<<<768, 256, 0, stream>>>(outputs, Wa_w, Wa_b, Ua_w, Ua_b, Wq, Uk);

  // Kernel 2: one block per (b, 4-row j tile): 8 * 128 = 1024 blocks
  score_softmax<<<Bdim * (Sdim / JT), 256, 0, stream>>>(Wq, Uk, Va_w, Va_b, mask, attnOut);

  // Kernel 3: 8 * (512/32)*(768/32) = 3072 waves -> 384 blocks
  context_gemm<<<384, 256, 0, stream>>>(attnOut, outputs, mask, ctxOut);
}